// L3GraphConv_84859963834407
// MI455X (gfx1250) — compile-verified
//
#include <hip/hip_runtime.h>
#include <hip/hip_bf16.h>

typedef __attribute__((ext_vector_type(2))) float v2f;
typedef __attribute__((ext_vector_type(8))) float v8f;

#define NNODES 20000

// ---------------------------------------------------------------------------
// Zero-fill a float buffer (grid-stride).
// ---------------------------------------------------------------------------
__global__ void zero_f32_kernel(float* __restrict__ p, int n) {
    int i = blockIdx.x * blockDim.x + threadIdx.x;
    int stride = gridDim.x * blockDim.x;
    for (; i < n; i += stride) p[i] = 0.0f;
}

// ---------------------------------------------------------------------------
// Zero-padded copy: dst[i] = (i < nReal) ? src[i] : 0  for i in [0, nPad).
// Rows are contiguous, so row-padding a [Nreal x K] matrix to [Npad x K] is
// exactly this 1-D predicate (no division needed).
// ---------------------------------------------------------------------------
__global__ void pad_copy_kernel(const float* __restrict__ src,
                                float* __restrict__ dst,
                                int nReal, int nPad) {
    int i = blockIdx.x * blockDim.x + threadIdx.x;
    int stride = gridDim.x * blockDim.x;
    for (; i < nPad; i += stride) dst[i] = (i < nReal) ? src[i] : 0.0f;
}

// ---------------------------------------------------------------------------
// Edge scatter-add: agg[dst[e]][f] += x[src[e]][f], width = 4*CHUNKS floats.
// CHUNKS is compile-time so the edge/chunk split is a mulhi, not a div.
// Consecutive threads walk chunks of the same edge -> coalesced float4 loads;
// x (<= 20.5 MB) lives in the 192 MB L2, so gather/scatter is L2 traffic.
// ---------------------------------------------------------------------------
template <int CHUNKS>
__global__ void edge_scatter_kernel(const float* __restrict__ x,
                                    float* __restrict__ agg,
                                    const int* __restrict__ src,
                                    const int* __restrict__ dst,
                                    int E) {
    const int total = E * CHUNKS;
    int i = blockIdx.x * blockDim.x + threadIdx.x;
    int stride = gridDim.x * blockDim.x;
    for (; i < total; i += stride) {
        const int e = i / CHUNKS;          // constant divisor -> mulhi
        const int c = i - e * CHUNKS;
        const int s = src[e];
        const int d = dst[e];
        const float4 v = *(const float4*)(x + (long)s * (4 * CHUNKS) + (c << 2));
        float* o = agg + (long)d * (4 * CHUNKS) + (c << 2);
        atomicAdd(o + 0, v.x);
        atomicAdd(o + 1, v.y);
        atomicAdd(o + 2, v.z);
        atomicAdd(o + 3, v.w);
    }
}

// ---------------------------------------------------------------------------
// Fused dual-GEMM + bias + ReLU using V_WMMA_F32_16X16X4_F32 (full fp32).
//   Out = relu(Agg @ Wrelp^T + X @ Wrootp^T + biasp)
// Wrelp/Wrootp are [Npad x K] zero-row-padded, biasp is [Npad] zero-padded,
// so the inner loop is branch-free: 4x global_load_b64 + 2x v_wmma per k-step.
// One wave per 16x16 C tile; 8 waves (256 threads) per block.
//
// f32 A 16x4 layout (ISA 7.12.2): lanes 0-15 hold rows M=0..15 with
//   VGPR0=K, VGPR1=K+1; lanes 16-31 hold K+2, K+3. B (4x16) mirrored.
// C: VGPR j, lanes 0-15 -> row j, col lane; lanes 16-31 -> row j+8.
// ---------------------------------------------------------------------------
__global__ void __launch_bounds__(256)
fused_gemm_relu_kernel(const float* __restrict__ Agg,    // [M x K]
                       const float* __restrict__ X,      // [M x K]
                       const float* __restrict__ Wrelp,  // [Npad x K], padded
                       const float* __restrict__ Wrootp, // [Npad x K], padded
                       const float* __restrict__ biasp,  // [Npad], padded
                       float* __restrict__ Out,          // [M x ldOut]
                       int M, int K, int Npad, int Nreal, int ldOut) {
    const int lane = threadIdx.x & 31;
    const int wave = threadIdx.x >> 5;
    const int tilesN = Npad >> 4;
    const int tilesM = M >> 4;               // M = 20000, multiple of 16
    const int tile = blockIdx.x * 8 + wave;  // wave-uniform
    if (tile >= tilesM * tilesN) return;     // uniform per wave: EXEC all-1s after

    const int tn = tile % tilesN;
    const int tm = tile / tilesN;
    const int laneM = lane & 15;
    const int kHalf = (lane >> 4) << 1;      // 0 for lanes 0-15, 2 for 16-31

    const int arow = (tm << 4) + laneM;      // A row this lane feeds
    const int ncol = (tn << 4) + laneM;      // output column this lane feeds

    const float* __restrict__ ap0 = Agg    + (long)arow * K + kHalf;
    const float* __restrict__ ap1 = X      + (long)arow * K + kHalf;
    const float* __restrict__ wp0 = Wrelp  + (long)ncol * K + kHalf;
    const float* __restrict__ wp1 = Wrootp + (long)ncol * K + kHalf;

    v8f acc = {};

#pragma unroll 4
    for (int k0 = 0; k0 < K; k0 += 4) {
        const v2f a0 = *(const v2f*)(ap0 + k0);   // 8B aligned (K, kHalf even)
        const v2f b0 = *(const v2f*)(wp0 + k0);
        const v2f a1 = *(const v2f*)(ap1 + k0);
        const v2f b1 = *(const v2f*)(wp1 + k0);
        acc = __builtin_amdgcn_wmma_f32_16x16x4_f32(
            false, a0, false, b0, (short)0, acc, false, false);
        acc = __builtin_amdgcn_wmma_f32_16x16x4_f32(
            false, a1, false, b1, (short)0, acc, false, false);
    }

    const float bn = biasp[ncol];            // padded -> unguarded
    const int rowBase = (tm << 4) + ((lane >> 4) << 3);
    if (ncol < Nreal) {                      // after all WMMAs; stores only
#pragma unroll
        for (int j = 0; j < 8; ++j) {
            float v = acc[j] + bn;
            v = v > 0.0f ? v : 0.0f;
            Out[(long)(rowBase + j) * ldOut + ncol] = v;
        }
    }
}

// ---------------------------------------------------------------------------
// Host-side orchestration.
// ---------------------------------------------------------------------------
static inline void launch_gemm(const float* agg, const float* x,
                               const float* wrelp, const float* wrootp,
                               const float* biasp, float* out,
                               int K, int Npad, int Nreal, int ldOut,
                               hipStream_t stream) {
    const int tiles = (NNODES >> 4) * (Npad >> 4);
    const int blocks = (tiles + 7) / 8;
    fused_gemm_relu_kernel<<<blocks, 256, 0, stream>>>(
        agg, x, wrelp, wrootp, biasp, out, NNODES, K, Npad, Nreal, ldOut);
}

static inline void launch_zero(float* p, int n, hipStream_t stream) {
    int blocks = (n + 255) / 256;
    if (blocks > 4096) blocks = 4096;
    zero_f32_kernel<<<blocks, 256, 0, stream>>>(p, n);
}

static inline void launch_pad(const float* src, float* dst, int nReal, int nPad,
                              hipStream_t stream) {
    int blocks = (nPad + 255) / 256;
    pad_copy_kernel<<<blocks, 256, 0, stream>>>(src, dst, nReal, nPad);
}

extern "C" void kernel_launch(void* const* d_in, const int* in_sizes, int n_in,
                              void* d_out, int out_size, void* d_ws, size_t ws_size,
                              hipStream_t stream) {
    const float* x       = (const float*)d_in[0];
    const int*   edge    = (const int*)d_in[1];
    const float* w_rel1  = (const float*)d_in[2];
    const float* b1      = (const float*)d_in[3];
    const float* w_root1 = (const float*)d_in[4];
    const float* w_rel2  = (const float*)d_in[5];
    const float* b2      = (const float*)d_in[6];
    const float* w_root2 = (const float*)d_in[7];
    const float* w_rel3  = (const float*)d_in[8];
    const float* b3      = (const float*)d_in[9];
    const float* w_root3 = (const float*)d_in[10];

    const int E = in_sizes[1] / 2;
    const int* src = edge;       // edge_index[0]
    const int* dst = edge + E;   // edge_index[1]

    // Layer dims: K (input width, real, mult of 4), Nreal -> Npad (mult of 16)
    // L1: 256 -> 200 (208)   L2: 200 -> 100 (112)   L3: 100 -> 2 (16)

    // Workspace layout (floats)
    float* agg  = (float*)d_ws;                       // 20000*256 (max agg)
    float* h1   = agg  + (size_t)NNODES * 256;        // 20000*200
    float* h2   = h1   + (size_t)NNODES * 200;        // 20000*100
    float* w1r  = h2   + (size_t)NNODES * 100;        // 208*256
    float* w1t  = w1r  + 208 * 256;                   // 208*256
    float* b1p  = w1t  + 208 * 256;                   // 208
    float* w2r  = b1p  + 208;                         // 112*200
    float* w2t  = w2r  + 112 * 200;                   // 112*200
    float* b2p  = w2t  + 112 * 200;                   // 112
    float* w3r  = b2p  + 112;                         // 16*100
    float* w3t  = w3r  + 16 * 100;                    // 16*100
    float* b3p  = w3t  + 16 * 100;                    // 16
    float* out  = (float*)d_out;                      // 20000*2

    // Pad weights/biases once per launch (tiny).
    launch_pad(w_rel1,  w1r, 200 * 256, 208 * 256, stream);
    launch_pad(w_root1, w1t, 200 * 256, 208 * 256, stream);
    launch_pad(b1,      b1p, 200,       208,       stream);
    launch_pad(w_rel2,  w2r, 100 * 200, 112 * 200, stream);
    launch_pad(w_root2, w2t, 100 * 200, 112 * 200, stream);
    launch_pad(b2,      b2p, 100,       112,       stream);
    launch_pad(w_rel3,  w3r, 2 * 100,   16 * 100,  stream);
    launch_pad(w_root3, w3t, 2 * 100,   16 * 100,  stream);
    launch_pad(b3,      b3p, 2,         16,        stream);

    // ---- Layer 1: x[20000x256] -> h1[20000x200] ----
    launch_zero(agg, NNODES * 256, stream);
    {
        int total = E * 64;
        int blocks = (total + 255) / 256; if (blocks > 65536) blocks = 65536;
        edge_scatter_kernel<64><<<blocks, 256, 0, stream>>>(x, agg, src, dst, E);
    }
    launch_gemm(agg, x, w1r, w1t, b1p, h1, /*K=*/256, /*Npad=*/208, /*Nreal=*/200,
                /*ldOut=*/200, stream);

    // ---- Layer 2: h1[20000x200] -> h2[20000x100] ----
    launch_zero(agg, NNODES * 200, stream);
    {
        int total = E * 50;
        int blocks = (total + 255) / 256; if (blocks > 65536) blocks = 65536;
        edge_scatter_kernel<50><<<blocks, 256, 0, stream>>>(h1, agg, src, dst, E);
    }
    launch_gemm(agg, h1, w2r, w2t, b2p, h2, /*K=*/200, /*Npad=*/112, /*Nreal=*/100,
                /*ldOut=*/100, stream);

    // ---- Layer 3: h2[20000x100] -> out[20000x2] ----
    launch_zero(agg, NNODES * 100, stream);
    {
        int total = E * 25;
        int blocks = (total + 255) / 256; if (blocks > 65536) blocks = 65536;
        edge_scatter_kernel<25><<<blocks, 256, 0, stream>>>(h2, agg, src, dst, E);
    }
    launch_gemm(agg, h2, w3r, w3t, b3p, out, /*K=*/100, /*Npad=*/16, /*Nreal=*/2,
                /*ldOut=*/2, stream);
}